// VKFF_46127948759311
// MI455X (gfx1250) — compile-verified
//
#include <hip/hip_runtime.h>
#include <hip/hip_bf16.h>

// ---------------------------------------------------------------------------
// Problem dimensions (from reference)
// ---------------------------------------------------------------------------
#define D_MODEL 2048
#define NH      8
#define DKH     256         // D_MODEL / H
#define D_INNER 4096
#define D_STATE 16
#define D_CONV  4
#define DT_RANK 128
#define B_SZ    8
#define LSEQ    1024
#define BL      (B_SZ * LSEQ)        // 8192 rows

typedef __attribute__((ext_vector_type(2))) float v2f;
typedef __attribute__((ext_vector_type(8))) float v8f;

__device__ __forceinline__ v8f wmma_f32(v2f a, v2f b, v8f c) {
  // D = A(16x4) * B(4x16) + C(16x16), fp32 matrix pipe (CDNA5)
  return __builtin_amdgcn_wmma_f32_16x16x4_f32(
      /*neg_a=*/false, a, /*neg_b=*/false, b,
      /*c_mod=*/(short)0, c, /*reuse_a=*/false, /*reuse_b=*/false);
}

__device__ __forceinline__ v8f v8zero() {
  v8f z = {0.f, 0.f, 0.f, 0.f, 0.f, 0.f, 0.f, 0.f};
  return z;
}

// ---------------------------------------------------------------------------
// fp32 WMMA GEMM:  C[M,N] = act(A[M,K] @ B[K,N] + bias)
// 256 threads = 8 waves; 128x128 block tile; each wave owns a 32x64 strip
// (2x4 WMMA tiles -> 8 v8f accumulators). K staged in 32-wide steps through
// double-buffered LDS: per step, 2 A-frag + 4 B-frag LDS reads feed 8 WMMAs.
// Global loads for step k+1 are staged in registers while computing step k.
// act: 0=none, 1=softplus, 2=silu
// ---------------------------------------------------------------------------
#define GBM 128
#define GBN 128
#define GBK 32
#define ASTR (GBK + 2)      // 34  (A tile row stride, floats)
#define BSTR (GBN + 4)      // 132 (B tile row stride, floats)
#define GEMM_LDS_FLOATS (2 * GBM * ASTR + 2 * GBK * BSTR)

__global__ __launch_bounds__(256, 1)
void gemm_f32_wmma(const float* __restrict__ A, int lda,
                   const float* __restrict__ B, int ldb,
                   const float* __restrict__ bias,
                   float* __restrict__ C, int ldc,
                   int M, int N, int K, int act)
{
  extern __shared__ float gsm[];
  float* Asm = gsm;                       // 2 * 128 * 34
  float* Bsm = gsm + 2 * GBM * ASTR;      // 2 * 32 * 132

  const int tid  = threadIdx.x;
  const int lane = tid & 31;
  const int wave = tid >> 5;              // 0..7
  const int wm   = (wave >> 1) * 32;      // wave row origin (0,32,64,96)
  const int wn   = (wave & 1) * 64;       // wave col origin (0,64)

  const int m0 = blockIdx.y * GBM;
  const int n0 = blockIdx.x * GBN;

  const int lm  = lane & 15;              // M / N index within fragment
  const int kb2 = (lane >> 4) * 2;        // K sub-offset (lanes 16..31 -> K+2)
  const int rh8 = (lane >> 4) * 8;        // C/D row-half offset

  const bool nfull = (n0 + GBN <= N);

  float4 ra[4], rb[4];

  auto loadA = [&](int k0) {
    #pragma unroll
    for (int i = 0; i < 4; ++i) {
      const int lin = tid + i * 256;          // 0..1023 float4s
      const int row = lin >> 3;               // 128 rows
      const int c4  = (lin & 7) * 4;          // 8 float4 per row
      ra[i] = *(const float4*)&A[(size_t)(m0 + row) * lda + (k0 + c4)];
    }
  };
  auto loadB = [&](int k0) {
    #pragma unroll
    for (int i = 0; i < 4; ++i) {
      const int lin = tid + i * 256;
      const int row = lin >> 5;               // 32 rows
      const int c4  = (lin & 31) * 4;         // 32 float4 per row
      const int gc  = n0 + c4;
      const float* bp = &B[(size_t)(k0 + row) * ldb + gc];
      if (nfull) {
        rb[i] = *(const float4*)bp;
      } else {
        rb[i].x = (gc + 0 < N) ? bp[0] : 0.f;
        rb[i].y = (gc + 1 < N) ? bp[1] : 0.f;
        rb[i].z = (gc + 2 < N) ? bp[2] : 0.f;
        rb[i].w = (gc + 3 < N) ? bp[3] : 0.f;
      }
    }
  };
  auto storeA = [&](int buf) {
    float* base = Asm + buf * GBM * ASTR;
    #pragma unroll
    for (int i = 0; i < 4; ++i) {
      const int lin = tid + i * 256;
      const int row = lin >> 3;
      const int c4  = (lin & 7) * 4;
      float* p = base + row * ASTR + c4;
      p[0] = ra[i].x; p[1] = ra[i].y; p[2] = ra[i].z; p[3] = ra[i].w;
    }
  };
  auto storeB = [&](int buf) {
    float* base = Bsm + buf * GBK * BSTR;
    #pragma unroll
    for (int i = 0; i < 4; ++i) {
      const int lin = tid + i * 256;
      const int row = lin >> 5;
      const int c4  = (lin & 31) * 4;
      float* p = base + row * BSTR + c4;
      p[0] = rb[i].x; p[1] = rb[i].y; p[2] = rb[i].z; p[3] = rb[i].w;
    }
  };

  v8f acc[8];
  #pragma unroll
  for (int i = 0; i < 8; ++i) acc[i] = v8zero();

  // prologue: stage tile 0
  loadA(0); loadB(0);
  storeA(0); storeB(0);
  __syncthreads();

  int cur = 0;
  for (int k0 = 0; k0 < K; k0 += GBK) {
    const bool more = (k0 + GBK < K);
    if (more) { loadA(k0 + GBK); loadB(k0 + GBK); }

    const float* Ab = Asm + cur * GBM * ASTR;
    const float* Bb = Bsm + cur * GBK * BSTR;
    #pragma unroll
    for (int kk = 0; kk < GBK; kk += 4) {
      v2f a0, a1, bb[4];
      a0.x = Ab[(wm + lm) * ASTR + kk + kb2];
      a0.y = Ab[(wm + lm) * ASTR + kk + kb2 + 1];
      a1.x = Ab[(wm + 16 + lm) * ASTR + kk + kb2];
      a1.y = Ab[(wm + 16 + lm) * ASTR + kk + kb2 + 1];
      #pragma unroll
      for (int j = 0; j < 4; ++j) {
        bb[j].x = Bb[(kk + kb2) * BSTR + wn + j * 16 + lm];
        bb[j].y = Bb[(kk + kb2 + 1) * BSTR + wn + j * 16 + lm];
      }
      #pragma unroll
      for (int j = 0; j < 4; ++j) {
        acc[j]     = wmma_f32(a0, bb[j], acc[j]);
        acc[4 + j] = wmma_f32(a1, bb[j], acc[4 + j]);
      }
    }

    if (more) {
      storeA(cur ^ 1); storeB(cur ^ 1);
      __syncthreads();
      cur ^= 1;
    }
  }

  // epilogue
  #pragma unroll
  for (int i = 0; i < 2; ++i) {
    #pragma unroll
    for (int j = 0; j < 4; ++j) {
      const v8f& t = acc[i * 4 + j];
      #pragma unroll
      for (int r = 0; r < 8; ++r) {
        const int grow = m0 + wm + i * 16 + r + rh8;
        const int gcol = n0 + wn + j * 16 + lm;
        if (grow < M && gcol < N) {
          float v = t[r];
          if (bias) v += bias[gcol];
          if (act == 1) v = (v > 20.f) ? v : log1pf(__expf(v));     // softplus
          else if (act == 2) v = v / (1.f + __expf(-v));            // silu
          C[(size_t)grow * ldc + gcol] = v;
        }
      }
    }
  }
}

// ---------------------------------------------------------------------------
// Flash attention: per (b,h) / 64-query block. Q,K,V are [B,L,H*DK] fp32.
// 256 threads = 8 waves. QK^T and P@V via WMMA f32 16x16x4 with fragment
// reuse (shared Q-frag across S tiles; hoisted P/V frags in the O update).
// ---------------------------------------------------------------------------
#define ATT_BQ 64
#define ATT_BK 64
#define QP (DKH + 4)        // 260, padded row stride for Q/K/V tiles
#define SP (ATT_BK + 4)     // 68, padded row stride for score tile

__global__ __launch_bounds__(256, 1)
void attn_kernel(const float* __restrict__ Q, const float* __restrict__ Kg,
                 const float* __restrict__ Vg, float* __restrict__ Oc)
{
  extern __shared__ float smem[];
  float* Qs   = smem;                       // 64*QP
  float* Ks   = Qs + 64 * QP;               // 64*QP
  float* Vs   = Ks + 64 * QP;               // 64*QP
  float* Sb   = Vs + 64 * QP;               // 64*SP
  float* red  = Sb + 64 * SP;               // 64*4
  float* mrow = red + 256;                  // 64
  float* lrow = mrow + 64;                  // 64
  float* srow = lrow + 64;                  // 64

  const int tid  = threadIdx.x;
  const int lane = tid & 31;
  const int wave = tid >> 5;                // 0..7
  const int lm   = lane & 15;
  const int kb2  = (lane >> 4) * 2;
  const int rh8  = (lane >> 4) * 8;

  const int bh = blockIdx.y;                // b*H + h
  const int b  = bh >> 3;
  const int h  = bh & 7;
  const int q0 = blockIdx.x * ATT_BQ;

  const size_t rs = (size_t)NH * DKH;       // 2048 row stride
  const float qscale = 0.0625f;             // 1/sqrt(256)

  // load Q tile (pre-scaled)
  for (int i = tid; i < 64 * (DKH / 4); i += 256) {
    const int r  = i >> 6;
    const int c4 = (i & 63) * 4;
    const float4 v = *(const float4*)&Q[(size_t)(b * LSEQ + q0 + r) * rs + h * DKH + c4];
    Qs[r * QP + c4 + 0] = v.x * qscale;
    Qs[r * QP + c4 + 1] = v.y * qscale;
    Qs[r * QP + c4 + 2] = v.z * qscale;
    Qs[r * QP + c4 + 3] = v.w * qscale;
  }
  if (tid < 64) { mrow[tid] = -1e30f; lrow[tid] = 0.f; }

  v8f o[8];
  #pragma unroll
  for (int i = 0; i < 8; ++i) o[i] = v8zero();

  for (int kt = 0; kt < LSEQ; kt += ATT_BK) {
    __syncthreads();
    // load K, V tiles
    for (int i = tid; i < 64 * 64; i += 256) {
      const int r  = i >> 6;
      const int c4 = (i & 63) * 4;
      const size_t g = (size_t)(b * LSEQ + kt + r) * rs + h * DKH + c4;
      const float4 kv = *(const float4*)&Kg[g];
      Ks[r * QP + c4 + 0] = kv.x;  Ks[r * QP + c4 + 1] = kv.y;
      Ks[r * QP + c4 + 2] = kv.z;  Ks[r * QP + c4 + 3] = kv.w;
      const float4 vv = *(const float4*)&Vg[g];
      Vs[r * QP + c4 + 0] = vv.x;  Vs[r * QP + c4 + 1] = vv.y;
      Vs[r * QP + c4 + 2] = vv.z;  Vs[r * QP + c4 + 3] = vv.w;
    }
    __syncthreads();

    // S = Qs @ Ks^T : wave owns rows smi..smi+15, cols snj0..snj0+31
    {
      const int smi  = (wave >> 1) * 16;
      const int snj0 = (wave & 1) * 32;
      v8f s0 = v8zero(), s1 = v8zero();
      #pragma unroll 4
      for (int d = 0; d < DKH; d += 4) {
        v2f a, b0, b1;
        a.x  = Qs[(smi + lm) * QP + d + kb2];
        a.y  = Qs[(smi + lm) * QP + d + kb2 + 1];
        b0.x = Ks[(snj0 + lm) * QP + d + kb2];
        b0.y = Ks[(snj0 + lm) * QP + d + kb2 + 1];
        b1.x = Ks[(snj0 + 16 + lm) * QP + d + kb2];
        b1.y = Ks[(snj0 + 16 + lm) * QP + d + kb2 + 1];
        s0 = wmma_f32(a, b0, s0);
        s1 = wmma_f32(a, b1, s1);
      }
      #pragma unroll
      for (int r = 0; r < 8; ++r) {
        Sb[(smi + r + rh8) * SP + snj0 + lm]      = s0[r];
        Sb[(smi + r + rh8) * SP + snj0 + 16 + lm] = s1[r];
      }
    }
    __syncthreads();

    // online softmax: 4 threads per row
    {
      const int rowi = tid >> 2, seg = tid & 3;
      float mx = -1e30f;
      #pragma unroll
      for (int j = 0; j < 16; ++j) mx = fmaxf(mx, Sb[rowi * SP + seg * 16 + j]);
      red[rowi * 4 + seg] = mx;
    }
    __syncthreads();
    if ((tid & 3) == 0) {
      const int rowi = tid >> 2;
      const float mx = fmaxf(fmaxf(red[rowi * 4], red[rowi * 4 + 1]),
                             fmaxf(red[rowi * 4 + 2], red[rowi * 4 + 3]));
      const float mo = mrow[rowi];
      const float mn = fmaxf(mo, mx);
      const float sc = __expf(mo - mn);
      mrow[rowi] = mn; srow[rowi] = sc; lrow[rowi] *= sc;
    }
    __syncthreads();
    {
      const int rowi = tid >> 2, seg = tid & 3;
      const float mn = mrow[rowi];
      float sum = 0.f;
      #pragma unroll
      for (int j = 0; j < 16; ++j) {
        const float p = __expf(Sb[rowi * SP + seg * 16 + j] - mn);
        Sb[rowi * SP + seg * 16 + j] = p;
        sum += p;
      }
      red[rowi * 4 + seg] = sum;
    }
    __syncthreads();
    if ((tid & 3) == 0) {
      const int rowi = tid >> 2;
      lrow[rowi] += red[rowi * 4] + red[rowi * 4 + 1] +
                    red[rowi * 4 + 2] + red[rowi * 4 + 3];
    }
    __syncthreads();

    // O = O*scale + P @ V ; wave owns O[:, wave*32 : wave*32+32] (8 tiles).
    #pragma unroll
    for (int tt2 = 0; tt2 < 8; ++tt2) {
      const int mi = (tt2 >> 1) * 16;
      #pragma unroll
      for (int r = 0; r < 8; ++r)
        o[tt2][r] *= srow[mi + r + rh8];
    }
    #pragma unroll
    for (int kk = 0; kk < ATT_BK; kk += 4) {
      v2f av[4], bv[2];
      #pragma unroll
      for (int mi_i = 0; mi_i < 4; ++mi_i) {
        av[mi_i].x = Sb[(mi_i * 16 + lm) * SP + kk + kb2];
        av[mi_i].y = Sb[(mi_i * 16 + lm) * SP + kk + kb2 + 1];
      }
      #pragma unroll
      for (int j = 0; j < 2; ++j) {
        const int nj = (wave * 2 + j) * 16;
        bv[j].x = Vs[(kk + kb2) * QP + nj + lm];
        bv[j].y = Vs[(kk + kb2 + 1) * QP + nj + lm];
      }
      #pragma unroll
      for (int mi_i = 0; mi_i < 4; ++mi_i)
        #pragma unroll
        for (int j = 0; j < 2; ++j)
          o[mi_i * 2 + j] = wmma_f32(av[mi_i], bv[j], o[mi_i * 2 + j]);
    }
  }
  __syncthreads();

  // write O / l
  #pragma unroll
  for (int tt2 = 0; tt2 < 8; ++tt2) {
    const int mi = (tt2 >> 1) * 16;
    const int nj = (wave * 2 + (tt2 & 1)) * 16;
    #pragma unroll
    for (int r = 0; r < 8; ++r) {
      const int row = mi + r + rh8;
      const float inv = 1.f / lrow[row];
      Oc[(size_t)(b * LSEQ + q0 + row) * rs + h * DKH + nj + lm] = o[tt2][r] * inv;
    }
  }
}

// ---------------------------------------------------------------------------
// Causal depthwise conv (D_CONV=4) + SiLU over x half of xz.
// ---------------------------------------------------------------------------
__global__ __launch_bounds__(256)
void conv_silu_kernel(const float* __restrict__ xz, const float* __restrict__ cw,
                      const float* __restrict__ cb, float* __restrict__ xc)
{
  const long idx = (long)blockIdx.x * 256 + threadIdx.x;
  if (idx >= (long)B_SZ * LSEQ * D_INNER) return;
  const int d = idx & (D_INNER - 1);
  const int t = (idx >> 12) & (LSEQ - 1);
  const int b = (int)(idx >> 22);
  float acc = cb[d];
  #pragma unroll
  for (int k = 0; k < D_CONV; ++k) {
    const int tt = t - (D_CONV - 1) + k;
    if (tt >= 0)
      acc += cw[d * D_CONV + k] * xz[(size_t)(b * LSEQ + tt) * (2 * D_INNER) + d];
  }
  xc[idx] = acc / (1.f + __expf(-acc));   // silu
}

// ---------------------------------------------------------------------------
// Selective scan: one thread per (b, d); 16-element state in registers.
// B_t / C_t broadcast via LDS. Reads x (xin) then writes y (yout) in place.
// ---------------------------------------------------------------------------
__global__ __launch_bounds__(256)
void scan_kernel(const float* xin, const float* __restrict__ dtb,
                 const float* __restrict__ dbc, const float* __restrict__ xzb,
                 const float* __restrict__ A_log, const float* __restrict__ Dskip,
                 float* yout)
{
  __shared__ float sBC[2 * D_STATE];
  const int d = blockIdx.x * 256 + threadIdx.x;   // 0..4095
  const int b = blockIdx.y;

  float Arow[D_STATE], h[D_STATE];
  #pragma unroll
  for (int n = 0; n < D_STATE; ++n) {
    Arow[n] = -__expf(A_log[d * D_STATE + n]);
    h[n] = 0.f;
  }
  const float Dv = Dskip[d];

  for (int t = 0; t < LSEQ; ++t) {
    __syncthreads();
    if (threadIdx.x < 2 * D_STATE)
      sBC[threadIdx.x] = dbc[(size_t)(b * LSEQ + t) * (DT_RANK + 2 * D_STATE) + DT_RANK + threadIdx.x];
    __syncthreads();

    const size_t ix = (size_t)(b * LSEQ + t) * D_INNER + d;
    const float xv  = xin[ix];
    const float dtv = dtb[ix];
    float y = 0.f;
    #pragma unroll
    for (int n = 0; n < D_STATE; ++n) {
      const float dA = __expf(dtv * Arow[n]);
      h[n] = dA * h[n] + (dtv * sBC[n]) * xv;
      y += h[n] * sBC[D_STATE + n];
    }
    y += Dv * xv;
    const float z = xzb[(size_t)(b * LSEQ + t) * (2 * D_INNER) + D_INNER + d];
    y *= z / (1.f + __expf(-z));
    yout[ix] = y;
  }
}

// ---------------------------------------------------------------------------
// Host launcher
// ---------------------------------------------------------------------------
extern "C" void kernel_launch(void* const* d_in, const int* in_sizes, int n_in,
                              void* d_out, int out_size, void* d_ws, size_t ws_size,
                              hipStream_t stream) {
  const float* x      = (const float*)d_in[0];
  const float* kf     = (const float*)d_in[1];
  const float* Wq     = (const float*)d_in[2];
  const float* bq     = (const float*)d_in[3];
  const float* Wk     = (const float*)d_in[4];
  const float* bk     = (const float*)d_in[5];
  const float* Wv     = (const float*)d_in[6];
  const float* bv     = (const float*)d_in[7];
  const float* Wo     = (const float*)d_in[8];
  const float* bo     = (const float*)d_in[9];
  const float* W_in   = (const float*)d_in[10];
  const float* conv_w = (const float*)d_in[11];
  const float* conv_b = (const float*)d_in[12];
  const float* W_ex   = (const float*)d_in[13];
  const float* W_xp   = (const float*)d_in[14];
  const float* W_dt   = (const float*)d_in[15];
  const float* b_dt   = (const float*)d_in[16];
  const float* A_log  = (const float*)d_in[17];
  const float* D_skip = (const float*)d_in[18];
  const float* W_out  = (const float*)d_in[19];

  float* out = (float*)d_out;
  float* ws  = (float*)d_ws;

  const size_t Mi = 1024 * 1024;
  float* qb   = ws;                  // 16Mi floats
  float* kb   = ws + 16 * Mi;        // 16Mi
  float* vb   = ws + 32 * Mi;        // 16Mi
  float* attc = ws + 48 * Mi;        // 16Mi
  // Mamba region (reuses MHA region after MHA completes on stream)
  float* xz   = ws;                  // 64Mi (8192 x 8192)
  float* xc   = ws + 64 * Mi;        // 32Mi (x_conv, later y)
  float* eb   = ws + 96 * Mi;        // 32Mi (e, later dt)
  float* dbc  = ws + 128 * Mi;       // 8192*160

  const size_t gemmSmem = (size_t)GEMM_LDS_FLOATS * sizeof(float);
  dim3 gblk(256);
  auto gemm = [&](const float* A, int lda, const float* B, int ldb,
                  const float* bias, float* C, int ldc,
                  int M, int N, int K, int act) {
    dim3 grid((N + GBN - 1) / GBN, (M + GBM - 1) / GBM);
    gemm_f32_wmma<<<grid, gblk, gemmSmem, stream>>>(A, lda, B, ldb, bias, C, ldc, M, N, K, act);
  };

  // ---- MHA branch -> out[0] ----
  gemm(x, D_MODEL, Wq, D_MODEL, bq, qb, D_MODEL, BL, D_MODEL, D_MODEL, 0);
  gemm(x, D_MODEL, Wk, D_MODEL, bk, kb, D_MODEL, BL, D_MODEL, D_MODEL, 0);
  gemm(x, D_MODEL, Wv, D_MODEL, bv, vb, D_MODEL, BL, D_MODEL, D_MODEL, 0);

  const size_t attSmem = (size_t)(3 * 64 * QP + 64 * SP + 256 + 3 * 64) * sizeof(float);
  attn_kernel<<<dim3(LSEQ / ATT_BQ, B_SZ * NH), dim3(256), attSmem, stream>>>(qb, kb, vb, attc);

  gemm(attc, D_MODEL, Wo, D_MODEL, bo, out, D_MODEL, BL, D_MODEL, D_MODEL, 0);

  // ---- Mamba branch -> out[1] ----
  gemm(x, D_MODEL, W_in, 2 * D_INNER, nullptr, xz, 2 * D_INNER, BL, 2 * D_INNER, D_MODEL, 0);

  conv_silu_kernel<<<(B_SZ * LSEQ * D_INNER) / 256, 256, 0, stream>>>(xz, conv_w, conv_b, xc);

  gemm(kf, D_MODEL, W_ex, D_INNER, nullptr, eb, D_INNER, BL, D_INNER, D_MODEL, 0);
  gemm(eb, D_INNER, W_xp, DT_RANK + 2 * D_STATE, nullptr, dbc, DT_RANK + 2 * D_STATE,
       BL, DT_RANK + 2 * D_STATE, D_INNER, 0);
  // dt = softplus(dbc[:, :128] @ W_dt + b_dt)  (writes over eb)
  gemm(dbc, DT_RANK + 2 * D_STATE, W_dt, D_INNER, b_dt, eb, D_INNER,
       BL, D_INNER, DT_RANK, 1);

  scan_kernel<<<dim3(D_INNER / 256, B_SZ), 256, 0, stream>>>(xc, eb, dbc, xz, A_log, D_skip, xc);

  gemm(xc, D_INNER, W_out, D_MODEL, nullptr, out + (size_t)BL * D_MODEL, D_MODEL,
       BL, D_MODEL, D_INNER, 0);
}